// rkhs_ouroboros_70970039599234
// MI455X (gfx1250) — compile-verified
//
#include <hip/hip_runtime.h>
#include <math.h>

// ---------------- problem constants (from reference) ----------------
#define DMF    4        // model dim (== WMMA K for f32 16x16x4)
#define EDF    4
#define NST    16       // SSM state size
#define BATCH  2
#define LSEQ   32768
#define LP     (LSEQ-8)         // 32760 tokens per sequence
#define NT     (BATCH*LP)       // 65520 total tokens (multiple of 16!)
#define XPD    33               // DTR + 2*N
#define TAUF   1000.0f
#define CHUNK  512
#define NCHUNK ((LP+CHUNK-1)/CHUNK)   // 64
#define TOTCH  (BATCH*NCHUNK)         // 128
#define MC     32               // RBF centers
// record layout in `rec` (stride 40): [0..3]=xc  [4..7]=delta  [8..23]=B  [24..39]=C
#define RSTR   40
#define PFDIST 8                // scan prefetch distance (tokens)

typedef __attribute__((ext_vector_type(2))) float v2f;
typedef __attribute__((ext_vector_type(8))) float v8f;

__device__ __forceinline__ float siluf(float x)     { return x / (1.0f + __expf(-x)); }
__device__ __forceinline__ float softplusf(float x) { return (x > 20.f) ? x : log1pf(__expf(x)); }

// D = A(16x4) * B(4x16) + C, f32.  A lanes: 0-15 -> K=0,1 ; 16-31 -> K=2,3 (ISA 7.12.2)
__device__ __forceinline__ v8f wmma_f32_k4(v2f a, v2f b, v8f c) {
    return __builtin_amdgcn_wmma_f32_16x16x4_f32(false, a, false, b, (short)0, c, false, false);
}

// ---------------- preprocessing ----------------
__global__ void k_prep_z(const float* __restrict__ x, float* __restrict__ z) {
    int t = blockIdx.x * blockDim.x + threadIdx.x;
    if (t >= NT) return;
    int b = t / LP, i = t % LP;
    const float* xb = x + (size_t)b * LSEQ;
    const float c1 = 4.f/5.f, c2 = -1.f/5.f, c3 = 4.f/105.f, c4 = -1.f/280.f;
    float xd = c1*(xb[i+5]-xb[i+3]) + c2*(xb[i+6]-xb[i+2]) +
               c3*(xb[i+7]-xb[i+1]) + c4*(xb[i+8]-xb[i]);
    z[2*t]   = xb[i+4];
    z[2*t+1] = xd;
}

__global__ void k_prep_resid(const float* __restrict__ z, float* __restrict__ ro,
                             float* __restrict__ rg, float* __restrict__ rk) {
    int t = blockIdx.x * blockDim.x + threadIdx.x;
    if (t >= NT) return;
    int b = t / LP, i = t % LP;
    int tf = b * LP + (LP - 1 - i);         // time-flip within the sequence
    float4 v = make_float4(z[2*t], z[2*t+1], z[2*tf], z[2*tf+1]);
    ((float4*)ro)[t] = v;
    ((float4*)rg)[t] = v;
    ((float4*)rk)[t] = v;
}

// ---------------- per-layer: rmsnorm + in_proj (WMMA) -> xz[NT*8] ----------------
__global__ void k_inproj(const float* __restrict__ resid, const float* __restrict__ norm_w,
                         const float* __restrict__ in_proj, float* __restrict__ xz) {
    int wave = threadIdx.x >> 5, lane = threadIdx.x & 31;
    int tile = blockIdx.x * (blockDim.x >> 5) + wave;
    if (tile * 16 >= NT) return;                 // uniform per wave: EXEC full at WMMA
    int  m  = lane & 15;
    int  t  = tile * 16 + m;                     // always < NT (NT % 16 == 0)
    bool hi = lane >= 16;
    int  k0 = hi ? 2 : 0;

    float4 r = ((const float4*)resid)[t];        // global_load_b128
    float rs = rsqrtf(0.25f*(r.x*r.x + r.y*r.y + r.z*r.z + r.w*r.w) + 1e-5f);
    float u[4] = { r.x*rs*norm_w[0], r.y*rs*norm_w[1], r.z*rs*norm_w[2], r.w*rs*norm_w[3] };

    v2f a; a[0] = u[k0]; a[1] = u[k0+1];
    int n = m;                                   // output column 0..15 (8 valid)
    v2f b; b[0] = (n < 8) ? in_proj[k0*8 + n]     : 0.f;
           b[1] = (n < 8) ? in_proj[(k0+1)*8 + n] : 0.f;
    v8f c = {};
    v8f d = wmma_f32_k4(a, b, c);

    int rbase = tile * 16 + (hi ? 8 : 0);
    if (n < 8) {                                 // single hoisted exec mask, 8 plain stores
        #pragma unroll
        for (int j = 0; j < 8; j++) xz[8*(rbase + j) + n] = d[j];
    }
}

// ---------------- fused conv+SiLU (per-lane channels) + x_proj WMMA ----------------
__global__ void k_xproj(const float* __restrict__ xz, const float* __restrict__ conv_w,
                        const float* __restrict__ conv_b, const float* __restrict__ x_proj,
                        float* __restrict__ rec, float* __restrict__ dbc) {
    int wave = threadIdx.x >> 5, lane = threadIdx.x & 31;
    int tile = blockIdx.x * (blockDim.x >> 5) + wave;
    if (tile * 16 >= NT) return;
    int  m  = lane & 15;
    int  t  = tile * 16 + m;
    bool hi = lane >= 16;
    int  k0 = hi ? 2 : 0;
    int  i  = t % LP;

    // causal depthwise conv over xz[.., 0..3]; lanes (m, m+16) cover all 4 channels of token t
    v2f a;
    #pragma unroll
    for (int q = 0; q < 2; q++) {
        int cc = k0 + q;
        float acc = conv_b[cc];
        #pragma unroll
        for (int k = 0; k < 4; k++) {
            int s = i - 3 + k;
            if (s >= 0) acc += xz[8*(t-3+k) + cc] * conv_w[cc*4 + k];
        }
        float xcv = siluf(acc);
        rec[(size_t)RSTR*t + cc] = xcv;          // xc needed by scan and out_proj
        a[q] = xcv;
    }

    int rbase = tile * 16 + (hi ? 8 : 0);
    #pragma unroll
    for (int nt = 0; nt < 3; nt++) {
        int col = nt * 16 + (lane & 15);
        v2f b; b[0] = (col < XPD) ? x_proj[k0*XPD + col]     : 0.f;
               b[1] = (col < XPD) ? x_proj[(k0+1)*XPD + col] : 0.f;
        v8f c = {};
        v8f d = wmma_f32_k4(a, b, c);
        if (col < XPD) {                         // constant-true for nt=0,1
            #pragma unroll
            for (int j = 0; j < 8; j++) dbc[(size_t)XPD*(rbase + j) + col] = d[j];
        }
    }
}

// ---------------- delta (softplus) + pack B,C into scan record ----------------
__global__ void k_deltabc(const float* __restrict__ dbc, const float* __restrict__ dtw,
                          const float* __restrict__ dtb, float* __restrict__ rec) {
    int t = blockIdx.x * blockDim.x + threadIdx.x;
    if (t >= NT) return;
    const float* s = dbc + (size_t)XPD*t;
    float* r = rec + (size_t)RSTR*t;
    float draw = s[0];
    #pragma unroll
    for (int e = 0; e < EDF; e++) r[4 + e] = softplusf(draw * dtw[e] + dtb[e]);
    #pragma unroll
    for (int n = 0; n < NST; n++) { r[8 + n] = s[1 + n]; r[24 + n] = s[17 + n]; }
}

// ---------------- chunked scan, phase 1: per-chunk (prodA, h_local) ----------------
// one wave per chunk; lane holds channels c=lane and c+32 (c = e*16+n)
__global__ void k_scan1(const float* __restrict__ rec, const float* __restrict__ A_log,
                        float* __restrict__ prodA, float* __restrict__ hloc) {
    int chunk = blockIdx.x, lane = threadIdx.x;
    int b = chunk / NCHUNK, ci = chunk % NCHUNK;
    int t0 = ci * CHUNK, t1 = t0 + CHUNK; if (t1 > LP) t1 = LP;
    int e0 = lane >> 4, n = lane & 15, e1 = 2 + (lane >> 4);
    float A0 = -__expf(A_log[lane]);
    float A1 = -__expf(A_log[lane + 32]);
    float h0 = 0.f, h1 = 0.f, p0 = 1.f, p1 = 1.f;
    for (int t = t0; t < t1; t++) {
        const float* r = rec + (size_t)RSTR * (b*LP + t);
        if (t + PFDIST < t1)                     // hide record latency in the serial chain
            __builtin_prefetch((const void*)(r + (size_t)RSTR*PFDIST), 0, 1);
        float d0 = r[4+e0], d1 = r[4+e1], Bn = r[8+n], x0 = r[e0], x1 = r[e1];
        float dA0 = __expf(d0 * A0), dA1 = __expf(d1 * A1);
        h0 = dA0*h0 + d0*Bn*x0;
        h1 = dA1*h1 + d1*Bn*x1;
        p0 *= dA0; p1 *= dA1;
    }
    prodA[64*chunk + lane] = p0;  prodA[64*chunk + lane + 32] = p1;
    hloc [64*chunk + lane] = h0;  hloc [64*chunk + lane + 32] = h1;
}

// ---------------- phase 2: sequential chunk-prefix combine (tiny) ----------------
__global__ void k_scan2(const float* __restrict__ prodA, const float* __restrict__ hloc,
                        float* __restrict__ pref) {
    int c = threadIdx.x;   // 64 threads = 64 channels
    for (int b = 0; b < BATCH; b++) {
        float carry = 0.f;
        for (int ci = 0; ci < NCHUNK; ci++) {
            int idx = b * NCHUNK + ci;
            pref[64*idx + c] = carry;
            carry = prodA[64*idx + c] * carry + hloc[64*idx + c];
        }
    }
}

// ---------------- phase 3: replay with prefix, fused einsum(h,C) -> yss ----------------
__global__ void k_scan3(const float* __restrict__ rec, const float* __restrict__ A_log,
                        const float* __restrict__ pref, float* __restrict__ yss) {
    int chunk = blockIdx.x, lane = threadIdx.x;
    int b = chunk / NCHUNK, ci = chunk % NCHUNK;
    int t0 = ci * CHUNK, t1 = t0 + CHUNK; if (t1 > LP) t1 = LP;
    int e0 = lane >> 4, n = lane & 15, e1 = 2 + (lane >> 4);
    float A0 = -__expf(A_log[lane]);
    float A1 = -__expf(A_log[lane + 32]);
    float h0 = pref[64*chunk + lane];
    float h1 = pref[64*chunk + lane + 32];
    for (int t = t0; t < t1; t++) {
        int tg = b*LP + t;
        const float* r = rec + (size_t)RSTR * tg;
        if (t + PFDIST < t1)
            __builtin_prefetch((const void*)(r + (size_t)RSTR*PFDIST), 0, 1);
        float d0 = r[4+e0], d1 = r[4+e1], Bn = r[8+n], x0 = r[e0], x1 = r[e1];
        float dA0 = __expf(d0 * A0), dA1 = __expf(d1 * A1);
        h0 = dA0*h0 + d0*Bn*x0;
        h1 = dA1*h1 + d1*Bn*x1;
        float Cn = r[24 + n];
        float v0 = h0 * Cn, v1 = h1 * Cn;
        #pragma unroll
        for (int msk = 1; msk < 16; msk <<= 1) {    // reduce over n within 16-lane groups
            v0 += __shfl_xor(v0, msk, 16);
            v1 += __shfl_xor(v1, msk, 16);
        }
        if (n == 0) {                                // lanes 0 and 16 hold the sums
            yss[4*tg + e0]     = v0;                 // e = 0 or 1
            yss[4*tg + 2 + e0] = v1;                 // e = 2 or 3
        }
    }
}

// ---------------- gate + D*xc + out_proj (WMMA) + residual add ----------------
__global__ void k_outproj(const float* __restrict__ yss, const float* __restrict__ rec,
                          const float* __restrict__ Dp, const float* __restrict__ xz,
                          const float* __restrict__ out_p, float* __restrict__ resid) {
    int wave = threadIdx.x >> 5, lane = threadIdx.x & 31;
    int tile = blockIdx.x * (blockDim.x >> 5) + wave;
    if (tile * 16 >= NT) return;
    int  m  = lane & 15;
    int  t  = tile * 16 + m;
    bool hi = lane >= 16;
    int  k0 = hi ? 2 : 0;

    const float* r = rec + (size_t)RSTR*t;
    v2f a;
    a[0] = (yss[4*t+k0]   + Dp[k0]  * r[k0])   * siluf(xz[8*t + 4 + k0]);
    a[1] = (yss[4*t+k0+1] + Dp[k0+1]* r[k0+1]) * siluf(xz[8*t + 4 + k0+1]);

    int n = lane & 15;
    v2f b; b[0] = (n < 4) ? out_p[k0*4 + n]     : 0.f;
           b[1] = (n < 4) ? out_p[(k0+1)*4 + n] : 0.f;
    v8f c = {};
    v8f d = wmma_f32_k4(a, b, c);
    int rbase = tile * 16 + (hi ? 8 : 0);
    if (n < 4) {                                 // single hoisted exec mask
        #pragma unroll
        for (int j = 0; j < 8; j++) resid[4*(rbase + j) + n] += d[j];
    }
}

// ---------------- heads: omega/gamma nets + RBF kernel ----------------
__global__ void k_heads(const float* __restrict__ ro, const float* __restrict__ rg,
                        const float* __restrict__ rk, const float* __restrict__ z,
                        const float* __restrict__ wo, const float* __restrict__ bo,
                        const float* __restrict__ wg, const float* __restrict__ bg,
                        const float* __restrict__ Wk, const float* __restrict__ cent,
                        const float* __restrict__ lsig,
                        float* __restrict__ om_raw, float* __restrict__ ga_raw,
                        float* __restrict__ rb_raw) {
    int t = blockIdx.x * blockDim.x + threadIdx.x;
    if (t >= NT) return;
    float4 oc = ((const float4*)ro)[t];
    float4 gc = ((const float4*)rg)[t];
    float4 kc = ((const float4*)rk)[t];
    float o = bo[0] + oc.x*wo[0] + oc.y*wo[1] + oc.z*wo[2] + oc.w*wo[3];
    float g = bg[0] + gc.x*wg[0] + gc.y*wg[1] + gc.z*wg[2] + gc.w*wg[3];
    om_raw[t] = fabsf(o);
    ga_raw[t] = g;
    float inv2s = __expf(-2.f * lsig[0]);
    float z0 = z[2*t], z1 = z[2*t+1];
    float s = 0.f;
    for (int j = 0; j < MC; j++) {
        float w = kc.x*Wk[0*MC+j] + kc.y*Wk[1*MC+j] + kc.z*Wk[2*MC+j] + kc.w*Wk[3*MC+j];
        float dx = z0 - cent[2*j], dy = z1 - cent[2*j+1];
        s += w * __expf(-0.5f * (dx*dx + dy*dy) * inv2s);
    }
    rb_raw[t] = s;
}

// ---------------- smooth (windowed mean, sw from dt) + yhat ----------------
__global__ void k_final(const float* __restrict__ om_raw, const float* __restrict__ ga_raw,
                        const float* __restrict__ rb_raw, const float* __restrict__ z,
                        const float* __restrict__ dtp, float* __restrict__ yhat) {
    int t = blockIdx.x * blockDim.x + threadIdx.x;
    if (t >= NT) return;
    int i = t % LP;
    float dt = dtp[0];
    int sw = (int)(0.001f / dt + 0.5f);
    if (sw < 1) sw = 1;
    if (sw > 4096) sw = 4096;
    int cnt = (i + 1 < sw) ? (i + 1) : sw;
    float so = 0.f, sg = 0.f, sr = 0.f;
    for (int k = 0; k < cnt; k++) { int tt = t - k; so += om_raw[tt]; sg += ga_raw[tt]; sr += rb_raw[tt]; }
    float inv = 1.f / (float)cnt;
    float om = so * inv, ga = (sg * inv) / TAUF, wk = (sr * inv) / TAUF;
    float z1 = z[2*t], z2 = z[2*t+1] / dt;
    yhat[t] = -om*om*z1 - ga*z2 - wk;
}

__global__ void k_copy_controls(const float* __restrict__ ro, const float* __restrict__ rg,
                                const float* __restrict__ rk, float* __restrict__ out) {
    int i = blockIdx.x * blockDim.x + threadIdx.x;
    const int one = NT * 4;
    if (i >= 3 * one) return;
    float v = (i < one) ? ro[i] : (i < 2*one) ? rg[i - one] : rk[i - 2*one];
    out[i] = v;
}

// ---------------- host orchestration ----------------
extern "C" void kernel_launch(void* const* d_in, const int* in_sizes, int n_in,
                              void* d_out, int out_size, void* d_ws, size_t ws_size,
                              hipStream_t stream) {
    (void)in_sizes; (void)n_in; (void)out_size; (void)ws_size;
    const float* x = (const float*)d_in[0];
    // layer fields (insertion order): norm_w,in_proj,conv_w,conv_b,x_proj,dt_w,dt_b,A_log,D,out_proj
    auto LPRM = [&](int s, int l, int f) { return (const float*)d_in[1 + (s*2 + l)*10 + f]; };
    const float* wo   = (const float*)d_in[61];
    const float* bo   = (const float*)d_in[62];
    const float* wg   = (const float*)d_in[63];
    const float* bg   = (const float*)d_in[64];
    const float* Wk   = (const float*)d_in[65];
    const float* cent = (const float*)d_in[66];
    const float* lsig = (const float*)d_in[67];
    const float* dtp  = (const float*)d_in[68];

    float* ws = (float*)d_ws;
    size_t off = 0;
    float* z      = ws + off; off += (size_t)NT*2;
    float* resid[3];
    for (int s = 0; s < 3; s++) { resid[s] = ws + off; off += (size_t)NT*4; }
    float* xz     = ws + off; off += (size_t)NT*8;       // [0..3]=pre-conv xc, [4..7]=z gate
    float* dbc    = ws + off; off += (size_t)NT*XPD;
    float* rec    = ws + off; off += (size_t)NT*RSTR;
    float* yss    = ws + off; off += (size_t)NT*4;
    float* prodA  = ws + off; off += (size_t)TOTCH*64;
    float* hloc   = ws + off; off += (size_t)TOTCH*64;
    float* pref   = ws + off; off += (size_t)TOTCH*64;
    float* om_raw = ws + off; off += NT;
    float* ga_raw = ws + off; off += NT;
    float* rb_raw = ws + off; off += NT;

    dim3 blk(256);
    int gTok  = (NT + 255) / 256;
    int tiles = NT / 16;             // exact: 4095
    int gTile = (tiles + 7) / 8;     // 8 waves (16-token tiles) per 256-thread block

    k_prep_z    <<<gTok, blk, 0, stream>>>(x, z);
    k_prep_resid<<<gTok, blk, 0, stream>>>(z, resid[0], resid[1], resid[2]);

    for (int s = 0; s < 3; s++) {
        for (int l = 0; l < 2; l++) {
            const float* norm_w = LPRM(s,l,0); const float* in_pr = LPRM(s,l,1);
            const float* conv_w = LPRM(s,l,2); const float* conv_b = LPRM(s,l,3);
            const float* x_pr   = LPRM(s,l,4); const float* dtw = LPRM(s,l,5);
            const float* dtb    = LPRM(s,l,6); const float* A_log = LPRM(s,l,7);
            const float* Dp     = LPRM(s,l,8); const float* out_p = LPRM(s,l,9);

            k_inproj  <<<gTile, blk, 0, stream>>>(resid[s], norm_w, in_pr, xz);
            k_xproj   <<<gTile, blk, 0, stream>>>(xz, conv_w, conv_b, x_pr, rec, dbc);
            k_deltabc <<<gTok,  blk, 0, stream>>>(dbc, dtw, dtb, rec);
            k_scan1   <<<TOTCH, 32, 0, stream>>>(rec, A_log, prodA, hloc);
            k_scan2   <<<1,     64, 0, stream>>>(prodA, hloc, pref);
            k_scan3   <<<TOTCH, 32, 0, stream>>>(rec, A_log, pref, yss);
            k_outproj <<<gTile, blk, 0, stream>>>(yss, rec, Dp, xz, out_p, resid[s]);
        }
    }

    k_heads<<<gTok, blk, 0, stream>>>(resid[0], resid[1], resid[2], z,
                                      wo, bo, wg, bg, Wk, cent, lsig,
                                      om_raw, ga_raw, rb_raw);
    k_final<<<gTok, blk, 0, stream>>>(om_raw, ga_raw, rb_raw, z, dtp, (float*)d_out);

    int ctot = 3 * NT * 4;
    k_copy_controls<<<(ctot + 255) / 256, blk, 0, stream>>>(resid[0], resid[1], resid[2],
                                                            ((float*)d_out) + NT);
}